// Rule_belief_layer_20804821582351
// MI455X (gfx1250) — compile-verified
//
#include <hip/hip_runtime.h>
#include <cstdint>

typedef __attribute__((ext_vector_type(2))) float v2f;
typedef __attribute__((ext_vector_type(4))) float v4f;
typedef __attribute__((ext_vector_type(8))) float v8f;

namespace {
constexpr int kB = 32768;
constexpr int kK = 64;
constexpr int kH = 64;
constexpr int kS = 68;                    // padded LDS row stride (floats) -> conflict-free
constexpr int kWaves = 8;                 // 256 threads = 8 wave32
constexpr int kRowsPerWave = 16;
constexpr int kRowsPerBlock = kWaves * kRowsPerWave;  // 128
}

static __device__ __forceinline__ v2f lo2(v4f v) { return __builtin_shufflevector(v, v, 0, 1); }
static __device__ __forceinline__ v2f hi2(v4f v) { return __builtin_shufflevector(v, v, 2, 3); }

__global__ __launch_bounds__(256, 1) void rule_belief_kernel(
    const float* __restrict__ W, const float* __restrict__ beta,
    float* __restrict__ out) {
  // ombT[n*kS + k] = 1 - beta[k*kH + n]  (transposed: k contiguous -> free v2f k-pairs)
  __shared__ __align__(16) float ombT[kH * kS];
  // per-wave W tile [16][kS]; reused afterwards as the p-tile buffer feeding WMMA A
  __shared__ __align__(16) float wp[kWaves][kRowsPerWave * kS];

  const int tid  = threadIdx.x;
  const int wave = tid >> 5;
  const int lane = tid & 31;
  const int h    = lane >> 4;   // which 16-lane half
  const int i16  = lane & 15;
  const int rowBase = blockIdx.x * kRowsPerBlock + wave * kRowsPerWave;

  // ---- stage (1 - beta) transposed (small, one-time) ----
  for (int idx = tid; idx < kK * kH; idx += 256) {
    const int k = idx >> 6;
    const int n = idx & 63;
    ombT[n * kS + k] = 1.0f - beta[idx];
  }

  // ---- stage this wave's 16x64 W tile via CDNA5 async global->LDS DMA ----
  float* wt = wp[wave];
#pragma unroll
  for (int r = 0; r < 8; ++r) {
    const int flat = r * 128 + lane * 4;     // 32 lanes * 16B each per pass
    const int m = flat >> 6;
    const int k = flat & 63;
    const unsigned ldsb = (unsigned)(uintptr_t)(wt + m * kS + k);
    const unsigned long long ga =
        (unsigned long long)(uintptr_t)(W + (size_t)(rowBase + m) * kK + k);
    asm volatile("global_load_async_to_lds_b128 %0, %1, off"
                 :: "v"(ldsb), "v"(ga) : "memory");
  }
  asm volatile("s_wait_asynccnt 0x0" ::: "memory");
  __syncthreads();

  // ---- semiring product over k, packed over k-pairs ----
  // accP[t][j] component 0 = product over even k's, component 1 = odd k's,
  // for output element p[j+8h][16t + i16].  acc2P likewise for term2.
  v2f accP[4][8];
  v2f acc2P[8];
#pragma unroll
  for (int j = 0; j < 8; ++j) {
    acc2P[j] = (v2f){1.0f, 1.0f};
#pragma unroll
    for (int t = 0; t < 4; ++t) accP[t][j] = (v2f){1.0f, 1.0f};
  }

  for (int k = 0; k < kK; k += 4) {
    v4f wv[8];
    v4f bv[4];
#pragma unroll
    for (int j = 0; j < 8; ++j)
      wv[j] = *reinterpret_cast<const v4f*>(wt + (j + 8 * h) * kS + k);
#pragma unroll
    for (int t = 0; t < 4; ++t)
      bv[t] = *reinterpret_cast<const v4f*>(&ombT[(16 * t + i16) * kS + k]);
#pragma unroll
    for (int u = 0; u < 2; ++u) {            // two k-pairs per 4-k chunk
      v2f ob2[4];
#pragma unroll
      for (int t = 0; t < 4; ++t) ob2[t] = u ? hi2(bv[t]) : lo2(bv[t]);
#pragma unroll
      for (int j = 0; j < 8; ++j) {
        const v2f w2 = u ? hi2(wv[j]) : lo2(wv[j]);      // free pair extract
        acc2P[j] = acc2P[j] - acc2P[j] * w2;             // -> v_pk_fma_f32
#pragma unroll
        for (int t = 0; t < 4; ++t) {
          const v2f f = 1.0f - w2 * ob2[t];              // -> v_pk_fma_f32 (neg)
          accP[t][j] *= f;                               // -> v_pk_mul_f32
        }
      }
    }
  }

  // ---- horizontal combine, p = term1 - term2; stash tile to LDS for WMMA A feed ----
  float acc[4][8];
#pragma unroll
  for (int j = 0; j < 8; ++j) {
    const float a2 = acc2P[j][0] * acc2P[j][1];
#pragma unroll
    for (int t = 0; t < 4; ++t) {
      acc[t][j] = accP[t][j][0] * accP[t][j][1] - a2;
      wt[(j + 8 * h) * kS + 16 * t + i16] = acc[t][j];
    }
  }

  // ---- rowsum[b] = P(16x64) x ones via 16 chained V_WMMA_F32_16X16X4_F32 ----
  // A chunk c: A[m][kk] = p[m][4c+kk]; lane (m = i16, half h) supplies K = {2h, 2h+1}
  // B = all-ones => D[m][n] = sum_kk A[m][kk], accumulated over chunks = full rowsum.
  v8f csum = {0.f, 0.f, 0.f, 0.f, 0.f, 0.f, 0.f, 0.f};
  const v2f bones = {1.0f, 1.0f};
#pragma unroll
  for (int c4 = 0; c4 < 16; ++c4) {
    const v2f a = *reinterpret_cast<const v2f*>(wt + i16 * kS + 4 * c4 + 2 * h);
    csum = __builtin_amdgcn_wmma_f32_16x16x4_f32(false, a, false, bones,
                                                 (short)0, csum, false, false);
  }

  // ---- normalize: C-layout of csum matches acc layout lane-for-lane ----
  // fast reciprocal (v_rcp_f32) + 1 Newton step instead of 8 full div chains
#pragma unroll
  for (int j = 0; j < 8; ++j) {
    const float s  = csum[j];                 // rowsum of row (j + 8h)
    const float r0 = __builtin_amdgcn_rcpf(s);
    const float e  = __builtin_fmaf(-s, r0, 1.0f);
    const float inv = __builtin_fmaf(r0, e, r0);
    const int row = rowBase + j + 8 * h;
#pragma unroll
    for (int t = 0; t < 4; ++t)
      out[row * kH + 16 * t + i16] = acc[t][j] * inv;
  }
}

extern "C" void kernel_launch(void* const* d_in, const int* in_sizes, int n_in,
                              void* d_out, int out_size, void* d_ws, size_t ws_size,
                              hipStream_t stream) {
  (void)in_sizes; (void)n_in; (void)d_ws; (void)ws_size; (void)out_size;
  const float* W    = (const float*)d_in[0];
  const float* beta = (const float*)d_in[1];
  float* out        = (float*)d_out;
  const int grid = kB / kRowsPerBlock;     // 256 blocks x 256 threads
  rule_belief_kernel<<<grid, 256, 0, stream>>>(W, beta, out);
}